// TA_Attention_42803644072167
// MI455X (gfx1250) — compile-verified
//
#include <hip/hip_runtime.h>
#include <math.h>

#define B_SIZE   2048
#define IN_DIM   512
#define QKV_DIM  1536     // 2*Q + V
#define N_HEADS  8
#define HEAD_DIM 64
#define VAL_DIM  512
#define TOPK     5

typedef float v2f __attribute__((ext_vector_type(2)));
typedef float v8f __attribute__((ext_vector_type(8)));

// D = A(16x4,f32) * B(4x16,f32) + C(16x16,f32)  -> v_wmma_f32_16x16x4_f32
__device__ __forceinline__ v8f wmma4(v2f a, v2f b, v8f c) {
  return __builtin_amdgcn_wmma_f32_16x16x4_f32(
      /*neg_a=*/false, a, /*neg_b=*/false, b,
      /*c_mod=*/(short)0, c, /*reuse_a=*/false, /*reuse_b=*/false);
}

// ---------------- 1) qkv = x @ W_qkv   [2048,512]x[512,1536] ----------------
__global__ void k_qkv_gemm(const float* __restrict__ x,
                           const float* __restrict__ w,
                           float* __restrict__ qkv) {
  const int lane = threadIdx.x & 31;
  const bool hi  = lane >= 16;
  const int l    = lane & 15;
  const int rowB = blockIdx.y * 16;
  const int colB = blockIdx.x * 16;
  const int ko   = hi ? 2 : 0;

  const float* ap = x + (rowB + l) * IN_DIM + ko;       // A row, dims ko..
  v8f acc = {0.f,0.f,0.f,0.f,0.f,0.f,0.f,0.f};
  for (int k0 = 0; k0 < IN_DIM; k0 += 4) {
    v2f a, b;
    a.x = ap[k0]; a.y = ap[k0 + 1];
    const float* bp = w + (size_t)(k0 + ko) * QKV_DIM + colB + l;
    b.x = bp[0];  b.y = bp[QKV_DIM];
    acc = wmma4(a, b, acc);
  }
  float* cp = qkv + colB + l;
#pragma unroll
  for (int r = 0; r < 8; ++r) {
    int row = rowB + r + (hi ? 8 : 0);
    cp[(size_t)row * QKV_DIM] = acc[r];
  }
}

// ---------------- 2) logits[h] = scale * q_h @ k_h^T  -> attn buffer --------
__global__ void k_logits(const float* __restrict__ qkv, float* __restrict__ attn) {
  const int lane = threadIdx.x & 31;
  const bool hi  = lane >= 16;
  const int l    = lane & 15;
  const int h    = blockIdx.z;
  const int qB   = blockIdx.y * 16;
  const int nB   = blockIdx.x * 16;
  const int ko   = hi ? 2 : 0;

  const float* qp = qkv + (size_t)(qB + l) * QKV_DIM + h * HEAD_DIM + ko;
  const float* kp = qkv + (size_t)(nB + l) * QKV_DIM + IN_DIM + h * HEAD_DIM + ko;
  v8f acc = {0.f,0.f,0.f,0.f,0.f,0.f,0.f,0.f};
  for (int k0 = 0; k0 < HEAD_DIM; k0 += 4) {
    v2f a, b;
    a.x = qp[k0]; a.y = qp[k0 + 1];   // q[qrow][k0+ko ..]
    b.x = kp[k0]; b.y = kp[k0 + 1];   // B[kk][n] = k[n][kk]
    acc = wmma4(a, b, acc);
  }
  const float scale = 0.125f;         // 1/sqrt(64)
  float* op = attn + (size_t)h * B_SIZE * B_SIZE + nB + l;
#pragma unroll
  for (int r = 0; r < 8; ++r) {
    int row = qB + r + (hi ? 8 : 0);
    op[(size_t)row * B_SIZE] = acc[r] * scale;
  }
}

// ---------------- 3) row softmax in place (one block per (h,q) row) ---------
__global__ void k_softmax(float* __restrict__ attn) {
  const int tid = threadIdx.x;               // 256 threads
  float* p = attn + (size_t)blockIdx.x * B_SIZE;
  __shared__ float red[256];
  float v[8];
  float vmax = -INFINITY;
#pragma unroll
  for (int i = 0; i < 8; ++i) { v[i] = p[tid + i * 256]; vmax = fmaxf(vmax, v[i]); }
  red[tid] = vmax; __syncthreads();
  for (int s = 128; s > 0; s >>= 1) {
    if (tid < s) red[tid] = fmaxf(red[tid], red[tid + s]);
    __syncthreads();
  }
  vmax = red[0]; __syncthreads();
  float sum = 0.f;
#pragma unroll
  for (int i = 0; i < 8; ++i) { v[i] = expf(v[i] - vmax); sum += v[i]; }
  red[tid] = sum; __syncthreads();
  for (int s = 128; s > 0; s >>= 1) {
    if (tid < s) red[tid] += red[tid + s];
    __syncthreads();
  }
  const float inv = 1.f / red[0];
#pragma unroll
  for (int i = 0; i < 8; ++i) p[tid + i * 256] = v[i] * inv;
}

// ---------------- 4) column sums per head -----------------------------------
__global__ void k_colsum(const float* __restrict__ attn, float* __restrict__ colsum) {
  const int h = blockIdx.y;
  const int c = blockIdx.x * 256 + threadIdx.x;
  const float* p = attn + (size_t)h * B_SIZE * B_SIZE + c;
  float s = 0.f;
  for (int q = 0; q < B_SIZE; ++q) s += p[(size_t)q * B_SIZE];
  colsum[h * B_SIZE + c] = s;
}

// ---------------- 5) top-5 argmax per head (iterative) ----------------------
__global__ void k_top5(const float* __restrict__ colsum, int* __restrict__ topidx) {
  const int h = blockIdx.x;
  const int tid = threadIdx.x;               // 256 threads
  __shared__ float vals[B_SIZE];
  __shared__ float rv[256];
  __shared__ int   ri[256];
  for (int i = tid; i < B_SIZE; i += 256) vals[i] = colsum[h * B_SIZE + i];
  __syncthreads();
  for (int t = 0; t < TOPK; ++t) {
    float bv = -INFINITY; int bi = 0;
    for (int i = tid; i < B_SIZE; i += 256) {
      float vv = vals[i];
      if (vv > bv) { bv = vv; bi = i; }      // earliest index on ties
    }
    rv[tid] = bv; ri[tid] = bi; __syncthreads();
    for (int s = 128; s > 0; s >>= 1) {
      if (tid < s) {
        if (rv[tid + s] > rv[tid] ||
            (rv[tid + s] == rv[tid] && ri[tid + s] < ri[tid])) {
          rv[tid] = rv[tid + s]; ri[tid] = ri[tid + s];
        }
      }
      __syncthreads();
    }
    if (tid == 0) { topidx[h * TOPK + t] = ri[0]; vals[ri[0]] = -INFINITY; }
    __syncthreads();
  }
}

// ---------------- 6) zero masked columns ------------------------------------
__global__ void k_mask(float* __restrict__ attn, const int* __restrict__ topidx) {
  const int h = blockIdx.x;
  const int tid = threadIdx.x;               // 256 threads
  for (int t = 0; t < TOPK; ++t) {
    const int c = topidx[h * TOPK + t];
    float* p = attn + (size_t)h * B_SIZE * B_SIZE + c;
    for (int q = tid; q < B_SIZE; q += 256) p[(size_t)q * B_SIZE] = 0.f;
  }
}

// ---------------- 7) ctx[b][h*64+n] = attn_h @ v_h  (K = 2048) --------------
__global__ void k_av_gemm(const float* __restrict__ attn,
                          const float* __restrict__ qkv,
                          float* __restrict__ ctx) {
  const int lane = threadIdx.x & 31;
  const bool hi  = lane >= 16;
  const int l    = lane & 15;
  const int h    = blockIdx.z;
  const int rowB = blockIdx.y * 16;
  const int nB   = blockIdx.x * 16;          // over HEAD_DIM=64 -> 4 tiles
  const int ko   = hi ? 2 : 0;

  const float* ap  = attn + (size_t)h * B_SIZE * B_SIZE
                   + (size_t)(rowB + l) * B_SIZE + ko;
  const float* bp0 = qkv + 2 * IN_DIM + h * HEAD_DIM + nB + l;   // v[kk][n]
  v8f acc = {0.f,0.f,0.f,0.f,0.f,0.f,0.f,0.f};
  for (int k0 = 0; k0 < B_SIZE; k0 += 4) {
    __builtin_prefetch(ap + k0 + 64, 0, 0);  // global_prefetch_b8, A stream
    v2f a, b;
    a.x = ap[k0]; a.y = ap[k0 + 1];
    const float* bp = bp0 + (size_t)(k0 + ko) * QKV_DIM;
    b.x = bp[0];  b.y = bp[QKV_DIM];
    acc = wmma4(a, b, acc);
  }
  float* cp = ctx + h * HEAD_DIM + nB + l;
#pragma unroll
  for (int r = 0; r < 8; ++r) {
    int row = rowB + r + (hi ? 8 : 0);
    cp[(size_t)row * VAL_DIM] = acc[r];
  }
}

// ---------------- 8) proj = ctx @ W_proj^T + b ------------------------------
__global__ void k_proj_gemm(const float* __restrict__ ctx,
                            const float* __restrict__ wproj,
                            const float* __restrict__ bproj,
                            float* __restrict__ proj) {
  const int lane = threadIdx.x & 31;
  const bool hi  = lane >= 16;
  const int l    = lane & 15;
  const int rowB = blockIdx.y * 16;
  const int colB = blockIdx.x * 16;
  const int ko   = hi ? 2 : 0;

  const float* ap = ctx + (size_t)(rowB + l) * VAL_DIM + ko;
  const float* bp = wproj + (size_t)(colB + l) * VAL_DIM + ko;  // W[o][kk], o=col
  v8f acc = {0.f,0.f,0.f,0.f,0.f,0.f,0.f,0.f};
  for (int k0 = 0; k0 < VAL_DIM; k0 += 4) {
    v2f a, b;
    a.x = ap[k0]; a.y = ap[k0 + 1];
    b.x = bp[k0]; b.y = bp[k0 + 1];
    acc = wmma4(a, b, acc);
  }
  const float bias = bproj[colB + l];
  float* cp = proj + colB + l;
#pragma unroll
  for (int r = 0; r < 8; ++r) {
    int row = rowB + r + (hi ? 8 : 0);
    cp[(size_t)row * VAL_DIM] = acc[r] + bias;
  }
}

// ---------------- 9) batch-norm stats (mean / rsqrt(var+eps)) ---------------
__global__ void k_bnstats(const float* __restrict__ proj,
                          float* __restrict__ mean, float* __restrict__ invstd) {
  const int c = blockIdx.x * 256 + threadIdx.x;
  float s = 0.f, s2 = 0.f;
  for (int b = 0; b < B_SIZE; ++b) {
    float v = proj[(size_t)b * VAL_DIM + c];
    s += v; s2 += v * v;
  }
  const float m   = s  * (1.f / B_SIZE);
  const float var = s2 * (1.f / B_SIZE) - m * m;
  mean[c]   = m;
  invstd[c] = rsqrtf(var + 1e-5f);
}

// ---------------- 10) BN apply + ReLU ---------------------------------------
__global__ void k_bnapply(const float* __restrict__ proj,
                          const float* __restrict__ mean,
                          const float* __restrict__ invstd,
                          const float* __restrict__ gamma,
                          const float* __restrict__ beta,
                          float* __restrict__ out) {
  const int i = blockIdx.x * 256 + threadIdx.x;
  const int c = i & (VAL_DIM - 1);
  const float y = gamma[c] * (proj[i] - mean[c]) * invstd[c] + beta[c];
  out[i] = fmaxf(y, 0.f);
}

extern "C" void kernel_launch(void* const* d_in, const int* in_sizes, int n_in,
                              void* d_out, int out_size, void* d_ws, size_t ws_size,
                              hipStream_t stream) {
  const float* x        = (const float*)d_in[0];   // [2048, 512]
  const float* W_qkv    = (const float*)d_in[1];   // [512, 1536]
  const float* W_proj   = (const float*)d_in[2];   // [512, 512]
  const float* b_proj   = (const float*)d_in[3];   // [512]
  const float* bn_gamma = (const float*)d_in[4];   // [512]
  const float* bn_beta  = (const float*)d_in[5];   // [512]

  float* out  = (float*)d_out;                                  // [2048,512]
  float* attn = (float*)d_out + (size_t)B_SIZE * VAL_DIM;       // [8,2048,2048]

  float* ws     = (float*)d_ws;
  float* qkv    = ws;                                           // 2048*1536
  float* ctx    = qkv + (size_t)B_SIZE * QKV_DIM;               // 2048*512
  float* proj   = ctx + (size_t)B_SIZE * VAL_DIM;               // 2048*512
  float* colsum = proj + (size_t)B_SIZE * VAL_DIM;              // 8*2048
  float* mean   = colsum + N_HEADS * B_SIZE;                    // 512
  float* invstd = mean + VAL_DIM;                               // 512
  int*   topidx = (int*)(invstd + VAL_DIM);                     // 8*5

  k_qkv_gemm <<<dim3(QKV_DIM/16, B_SIZE/16),          32,  0, stream>>>(x, W_qkv, qkv);
  k_logits   <<<dim3(B_SIZE/16, B_SIZE/16, N_HEADS),  32,  0, stream>>>(qkv, attn);
  k_softmax  <<<N_HEADS * B_SIZE,                     256, 0, stream>>>(attn);
  k_colsum   <<<dim3(B_SIZE/256, N_HEADS),            256, 0, stream>>>(attn, colsum);
  k_top5     <<<N_HEADS,                              256, 0, stream>>>(colsum, topidx);
  k_mask     <<<N_HEADS,                              256, 0, stream>>>(attn, topidx);
  k_av_gemm  <<<dim3(HEAD_DIM/16, B_SIZE/16, N_HEADS),32,  0, stream>>>(attn, qkv, ctx);
  k_proj_gemm<<<dim3(VAL_DIM/16, B_SIZE/16),          32,  0, stream>>>(ctx, W_proj, b_proj, proj);
  k_bnstats  <<<VAL_DIM/256,                          256, 0, stream>>>(proj, mean, invstd);
  k_bnapply  <<<(B_SIZE*VAL_DIM)/256,                 256, 0, stream>>>(proj, mean, invstd,
                                                                        bn_gamma, bn_beta, out);
}